// SGL_Encoder_16724602651074
// MI455X (gfx1250) — compile-verified
//
#include <hip/hip_runtime.h>
#include <hip/hip_bf16.h>
#include <stdint.h>
#include <stddef.h>

#ifndef __has_builtin
#define __has_builtin(x) 0
#endif

// gfx1250 async global->LDS copy path (ASYNCcnt); guarded so the file compiles
// on toolchains without the builtins (falls back to direct loads + prefetch).
#if __has_builtin(__builtin_amdgcn_global_load_async_to_lds_b32)
#define ASYNC_LDS 1
#else
#define ASYNC_LDS 0
#endif

#if __has_builtin(__builtin_amdgcn_s_wait_asynccnt)
#define WAIT_ASYNC(n) __builtin_amdgcn_s_wait_asynccnt(n)
#elif ASYNC_LDS
#define WAIT_ASYNC(n) asm volatile("s_wait_asynccnt %0" ::"i"(n) : "memory")
#else
#define WAIT_ASYNC(n)
#endif

// Typed address-space casts: the builtin's params are AS(1) int* / AS(3) int*.
#define GPTR_I32(p) ((__attribute__((address_space(1))) int*)(void*)(p))
#define LPTR_I32(p) ((__attribute__((address_space(3))) int*)(void*)(p))

#define BLOCK 256
#define WAVES_PER_BLOCK 8   // wave32
#define CHUNK 32            // edges staged per wave per iteration

// ---------------------------------------------------------------------------
// init: out = cur = concat(user, item); nxt = 0
// ---------------------------------------------------------------------------
__global__ __launch_bounds__(BLOCK) void init_k(
    const float4* __restrict__ user, const float4* __restrict__ item,
    float4* __restrict__ out, float4* __restrict__ cur, float4* __restrict__ nxt,
    int n4, int nUser4) {
  int i = blockIdx.x * blockDim.x + threadIdx.x;
  if (i >= n4) return;
  float4 v = (i < nUser4) ? user[i] : item[i - nUser4];
  out[i] = v;
  cur[i] = v;
  nxt[i] = make_float4(0.f, 0.f, 0.f, 0.f);
}

// ---------------------------------------------------------------------------
// accum: out = (out + nxt) * scale; also zero the buffer that becomes the
// next layer's scatter destination (saves a separate zeroing pass).
// ---------------------------------------------------------------------------
__global__ __launch_bounds__(BLOCK) void accum_k(
    float4* __restrict__ out, const float4* __restrict__ nxt,
    float4* __restrict__ zbuf, float scale, int n4) {
  int i = blockIdx.x * blockDim.x + threadIdx.x;
  if (i >= n4) return;
  float4 o = out[i];
  float4 x = nxt[i];
  o.x = (o.x + x.x) * scale;
  o.y = (o.y + x.y) * scale;
  o.z = (o.z + x.z) * scale;
  o.w = (o.w + x.w) * scale;
  out[i] = o;
  zbuf[i] = make_float4(0.f, 0.f, 0.f, 0.f);
}

// ---------------------------------------------------------------------------
// spmm: dst[rows[e]] += src[cols[e]] * vals[e] for all edges.
// 16 lanes per edge (one float4 per lane), 2 edges per wave32 step.
// Each wave grid-strides over 32-edge chunks; chunk index data is streamed
// into LDS with double-buffered async global->LDS loads when available.
// ---------------------------------------------------------------------------
__global__ __launch_bounds__(BLOCK) void spmm_k(
    const int* __restrict__ rows, const int* __restrict__ cols,
    const float* __restrict__ vals, const float* __restrict__ src,
    float* __restrict__ dst, int nnz) {
#if ASYNC_LDS
  __shared__ int smem[WAVES_PER_BLOCK][2][3 * CHUNK];
#endif
  const int wave = threadIdx.x >> 5;
  const int lane = threadIdx.x & 31;
  const int half = lane >> 4;   // which of the 2 edges this half-wave handles
  const int l16  = lane & 15;   // float4 slot within the 64-wide embedding

  const long nnzL    = (long)nnz;
  const long nchunks = (nnzL + CHUNK - 1) / CHUNK;
  const long stride  = (long)gridDim.x * WAVES_PER_BLOCK;
  long c = (long)blockIdx.x * WAVES_PER_BLOCK + wave;
  if (c >= nchunks) return;

#if ASYNC_LDS
  int curbuf = 0;
  {  // prime buffer 0
    long e = c * CHUNK + lane;
    if (e > nnzL - 1) e = nnzL - 1;
    __builtin_amdgcn_global_load_async_to_lds_b32(
        GPTR_I32(rows + e), LPTR_I32(&smem[wave][0][lane]), 0, 0);
    __builtin_amdgcn_global_load_async_to_lds_b32(
        GPTR_I32(cols + e), LPTR_I32(&smem[wave][0][CHUNK + lane]), 0, 0);
    __builtin_amdgcn_global_load_async_to_lds_b32(
        GPTR_I32(vals + e), LPTR_I32(&smem[wave][0][2 * CHUNK + lane]), 0, 0);
  }
#endif

  for (; c < nchunks; c += stride) {
    const long cn = c + stride;
#if ASYNC_LDS
    const int* bcur = smem[wave][curbuf];
    if (cn < nchunks) {  // stage next chunk into the other buffer
      int* bnext = smem[wave][curbuf ^ 1];
      long e = cn * CHUNK + lane;
      if (e > nnzL - 1) e = nnzL - 1;
      __builtin_amdgcn_global_load_async_to_lds_b32(
          GPTR_I32(rows + e), LPTR_I32(&bnext[lane]), 0, 0);
      __builtin_amdgcn_global_load_async_to_lds_b32(
          GPTR_I32(cols + e), LPTR_I32(&bnext[CHUNK + lane]), 0, 0);
      __builtin_amdgcn_global_load_async_to_lds_b32(
          GPTR_I32(vals + e), LPTR_I32(&bnext[2 * CHUNK + lane]), 0, 0);
      WAIT_ASYNC(3);  // loads complete in order: first 3 (current buffer) done
    } else {
      WAIT_ASYNC(0);
    }
#else
    if (cn < nchunks) {  // software prefetch of next chunk's index data
      __builtin_prefetch(rows + cn * CHUNK, 0, 0);
      __builtin_prefetch(cols + cn * CHUNK, 0, 0);
      __builtin_prefetch(vals + cn * CHUNK, 0, 0);
    }
#endif
    const long base = c * CHUNK;
#pragma unroll 4
    for (int k = 0; k < CHUNK / 2; ++k) {
      const int eL  = 2 * k + half;
      const long eg = base + eL;
      if (eg < nnzL) {
#if ASYNC_LDS
        const int   r  = bcur[eL];
        const int   ci = bcur[CHUNK + eL];
        const float v  = __int_as_float(bcur[2 * CHUNK + eL]);
#else
        const int   r  = rows[eg];
        const int   ci = cols[eg];
        const float v  = vals[eg];
#endif
        const float4 g = *(const float4*)(src + (size_t)ci * 64 + l16 * 4);
        float* d = dst + (size_t)r * 64 + l16 * 4;
        unsafeAtomicAdd(d + 0, g.x * v);
        unsafeAtomicAdd(d + 1, g.y * v);
        unsafeAtomicAdd(d + 2, g.z * v);
        unsafeAtomicAdd(d + 3, g.w * v);
      }
    }
#if ASYNC_LDS
    curbuf ^= 1;
#endif
  }
}

// ---------------------------------------------------------------------------
extern "C" void kernel_launch(void* const* d_in, const int* in_sizes, int n_in,
                              void* d_out, int out_size, void* d_ws, size_t ws_size,
                              hipStream_t stream) {
  const float* user = (const float*)d_in[0];   // 200000*64 f32
  const float* item = (const float*)d_in[1];   // 100000*64 f32
  const float* vals = (const float*)d_in[2];   // 12.8M f32
  const int*   rows = (const int*)d_in[3];     // 12.8M i32
  const int*   cols = (const int*)d_in[4];     // 12.8M i32

  const int nnz    = in_sizes[2];
  const int nUserF = in_sizes[0];
  const int nItemF = in_sizes[1];
  const int nF     = nUserF + nItemF;  // 19,200,000 floats
  const int n4     = nF / 4;
  const int nUser4 = nUserF / 4;

  float* out = (float*)d_out;
  float* ws0 = (float*)d_ws;        // ping-pong ego buffers (76.8 MB each)
  float* ws1 = ws0 + (size_t)nF;

  const int vb = (n4 + BLOCK - 1) / BLOCK;
  init_k<<<vb, BLOCK, 0, stream>>>((const float4*)user, (const float4*)item,
                                   (float4*)out, (float4*)ws0, (float4*)ws1,
                                   n4, nUser4);

  float* cur = ws0;
  float* nxt = ws1;
  const int SPMM_BLOCKS = 4096;  // 32768 waves grid-striding over 400K chunks
  for (int l = 0; l < 3; ++l) {
    spmm_k<<<SPMM_BLOCKS, BLOCK, 0, stream>>>(rows, cols, vals, cur, nxt, nnz);
    const float scale = (l == 2) ? 0.25f : 1.0f;
    accum_k<<<vb, BLOCK, 0, stream>>>((float4*)out, (const float4*)nxt,
                                      (float4*)cur, scale, n4);
    float* t = cur; cur = nxt; nxt = t;
  }
}